// WeightedPersistenceLayer_23313082483164
// MI455X (gfx1250) — compile-verified
//
#include <hip/hip_runtime.h>
#include <math.h>

// Shapes from the reference
#define BATCH   8
#define NPTS    1024
#define DIM     3
#define HID     32
#define NSCALES 5

typedef float v2f __attribute__((ext_vector_type(2)));
typedef float v8f __attribute__((ext_vector_type(8)));

// ---------------------------------------------------------------------------
// Prep: per-point squared norms, MLP filtration (halved), softplus scales,
// softmax importance (written to tail of d_out).
// ---------------------------------------------------------------------------
__global__ __launch_bounds__(256) void wpl_prep_kernel(
    const float* __restrict__ x,            // [B,N,3]
    const float* __restrict__ weights,      // [B,N]
    const float* __restrict__ W1,           // [1,32]
    const float* __restrict__ b1,           // [32]
    const float* __restrict__ W2,           // [32,1]
    const float* __restrict__ b2,           // [1]
    const float* __restrict__ scale_params, // [5]
    const float* __restrict__ logits,       // [5]
    float* __restrict__ fh,                 // [B*N] 0.5*f
    float* __restrict__ sq,                 // [B*N] |x|^2
    float* __restrict__ scales,             // [5]  softplus
    float* __restrict__ importance_out)     // [5]  softmax -> d_out tail
{
    const int p = blockIdx.x * blockDim.x + threadIdx.x;   // 0..8191
    const float w = weights[p];
    float acc = b2[0];
    #pragma unroll
    for (int k = 0; k < HID; ++k) {
        float h = fmaf(w, W1[k], b1[k]);
        h = fmaxf(h, 0.0f);
        acc = fmaf(h, W2[k], acc);
    }
    fh[p] = 0.5f * acc;

    const float x0 = x[3 * p + 0];
    const float x1 = x[3 * p + 1];
    const float x2 = x[3 * p + 2];
    sq[p] = x0 * x0 + x1 * x1 + x2 * x2;

    if (blockIdx.x == 0 && threadIdx.x == 0) {
        // softplus (stable): max(v,0) + log1p(exp(-|v|))
        #pragma unroll
        for (int s = 0; s < NSCALES; ++s) {
            float v = scale_params[s];
            scales[s] = fmaxf(v, 0.0f) + log1pf(expf(-fabsf(v)));
        }
        // softmax
        float l[NSCALES], m = -3.0e38f;
        #pragma unroll
        for (int s = 0; s < NSCALES; ++s) { l[s] = logits[s]; m = fmaxf(m, l[s]); }
        float e[NSCALES], sum = 0.0f;
        #pragma unroll
        for (int s = 0; s < NSCALES; ++s) { e[s] = expf(l[s] - m); sum += e[s]; }
        #pragma unroll
        for (int s = 0; s < NSCALES; ++s) importance_out[s] = e[s] / sum;
    }
}

// ---------------------------------------------------------------------------
// Main: one wave32 computes one 16x16 (i,j) tile of one batch via
// V_WMMA_F32_16X16X4_F32 (K padded 3->4), then writes 5 scaled copies.
// 32768 tiles total, 8 waves/block -> 4096 blocks, EXEC all-ones everywhere.
// ---------------------------------------------------------------------------
__global__ __launch_bounds__(256) void wpl_tile_kernel(
    const float* __restrict__ x,        // [B,N,3]
    const float* __restrict__ fh,       // [B*N]
    const float* __restrict__ sq,       // [B*N]
    const float* __restrict__ scales_p, // [5]
    float* __restrict__ out)            // [5,B,N,N]
{
    const int wave = threadIdx.x >> 5;
    const int lane = threadIdx.x & 31;

    const int t   = blockIdx.x * 8 + wave;   // tile id 0..32767
    const int b   = t >> 12;                 // / (64*64)
    const int rem = t & 4095;
    const int i0  = (rem >> 6) << 4;
    const int j0  = (rem & 63) << 4;

    const int half = lane >> 4;              // 0: K=0,1   1: K=2,3(pad)
    const int n    = lane & 15;              // row (A) / col (B) within tile
    const int k0   = half << 1;

    const float* xb = x + (size_t)b * NPTS * DIM;

    // A (16x4, M striped over lanes): lane holds x[i0+n][k0], x[i0+n][k0+1]
    v2f av, bv;
    {
        const float* pa = xb + (size_t)(i0 + n) * DIM;
        av.x = pa[k0];
        av.y = half ? 0.0f : pa[1];
        const float* pb = xb + (size_t)(j0 + n) * DIM;
        bv.x = pb[k0];
        bv.y = half ? 0.0f : pb[1];
    }

    v8f c = {0.f, 0.f, 0.f, 0.f, 0.f, 0.f, 0.f, 0.f};
    // D = A x B + C : 16x16 Gram tile (x_i . x_j)
    c = __builtin_amdgcn_wmma_f32_16x16x4_f32(
        /*neg_a=*/false, av, /*neg_b=*/false, bv,
        /*c_mod=*/(short)0, c, /*reuse_a=*/false, /*reuse_b=*/false);

    const int base = b * NPTS;
    const float sqj = sq[base + j0 + n];
    const float fhj = fh[base + j0 + n];
    const float s0 = scales_p[0], s1 = scales_p[1], s2 = scales_p[2],
                s3 = scales_p[3], s4 = scales_p[4];

    // C/D layout: VGPR r holds row M = r + 8*half, column N = lane&15
    const int irow0 = i0 + (half << 3);
    #pragma unroll
    for (int r = 0; r < 8; ++r) {
        const int i = irow0 + r;
        const float d2  = sq[base + i] + sqj - 2.0f * c[r];
        const float wgt = sqrtf(fmaxf(d2, 0.0f)) + fh[base + i] + fhj;
        const size_t off = ((size_t)b << 20) + ((size_t)i << 10) + (size_t)(j0 + n);
        out[off                        ] = s0 * wgt;
        out[off + ((size_t)1 << 23)    ] = s1 * wgt;   // B*N*N = 1<<23 per scale
        out[off + ((size_t)2 << 23)    ] = s2 * wgt;
        out[off + ((size_t)3 << 23)    ] = s3 * wgt;
        out[off + ((size_t)4 << 23)    ] = s4 * wgt;
    }
}

// ---------------------------------------------------------------------------
extern "C" void kernel_launch(void* const* d_in, const int* in_sizes, int n_in,
                              void* d_out, int out_size, void* d_ws, size_t ws_size,
                              hipStream_t stream) {
    const float* x            = (const float*)d_in[0];
    const float* weights      = (const float*)d_in[1];
    const float* W1           = (const float*)d_in[2];
    const float* b1           = (const float*)d_in[3];
    const float* W2           = (const float*)d_in[4];
    const float* b2           = (const float*)d_in[5];
    const float* scale_params = (const float*)d_in[6];
    const float* logits       = (const float*)d_in[7];

    float* out = (float*)d_out;
    float* fh     = (float*)d_ws;                  // 8192 floats
    float* sq     = fh + BATCH * NPTS;             // 8192 floats
    float* scales = sq + BATCH * NPTS;             // 5 floats

    float* importance_out = out + (size_t)NSCALES * BATCH * NPTS * NPTS;

    wpl_prep_kernel<<<(BATCH * NPTS) / 256, 256, 0, stream>>>(
        x, weights, W1, b1, W2, b2, scale_params, logits,
        fh, sq, scales, importance_out);

    // 8 batches * 64*64 tiles = 32768 tiles; 8 waves/block -> 4096 blocks
    wpl_tile_kernel<<<4096, 256, 0, stream>>>(x, fh, sq, scales, out);
}